// Earth4D_48610439856781
// MI455X (gfx1250) — compile-verified
//
#include <hip/hip_runtime.h>
#include <cstdint>
#include <cmath>

// ---------------------------------------------------------------------------
// Earth4D multiresolution hash encoding for MI455X (gfx1250, wave32).
// Gather-bound: 256M random 8B table gathers dominate; no matmul -> WMMA does
// not apply. CDNA5 paths used: Tensor Data Mover (tensor_load_to_lds) staging
// of coord tiles into LDS (OOB zero-fill handles the tail block),
// s_wait_tensorcnt, wave-uniform scalar table bases, factored corner hashing,
// b64 LDS staging + coalesced b128 output stores.
// ---------------------------------------------------------------------------

#define LVLS 16
#define TBL  524288            // 1 << 19
#define TMASK 524287u
#define PTS_PER_BLK 64
#define THREADS 256
#define SROW 130               // sOut row stride (dwords), even + conflict-free

typedef unsigned int u32;
typedef uint32_t u32x4 __attribute__((ext_vector_type(4)));
typedef int      i32x8 __attribute__((ext_vector_type(8)));
typedef int      i32x4 __attribute__((ext_vector_type(4)));

struct ResConst {
  int sres[LVLS];     // spatial res (same all 3 dims)
  int tres01[LVLS];   // temporal res dims 0,1
  int tres2[LVLS];    // temporal res dim 2
};

// Monotone float<->uint mapping so unsigned atomicMin/Max implement fp min/max
__device__ __forceinline__ u32 f2ord(float f) {
  u32 u = __float_as_uint(f);
  return (u & 0x80000000u) ? ~u : (u | 0x80000000u);
}
__device__ __forceinline__ float ord2f(u32 v) {
  u32 u = (v & 0x80000000u) ? (v ^ 0x80000000u) : ~v;
  return __uint_as_float(u);
}

__device__ __forceinline__ void ecef(float lat, float lon, float elev,
                                     float& cx, float& cy, float& cz) {
  const float D2R = 0.017453292519943295f;
  float latr = lat * D2R, lonr = lon * D2R;
  float r = 6371000.0f + elev;
  float cl = cosf(latr), sl = sinf(latr);
  float co = cosf(lonr), so = sinf(lonr);
  cx = r * cl * co;
  cy = r * cl * so;
  cz = r * sl;
}

// ws layout (8 uints): [0..3] = ordered-min of (x,y,z,t), [4..7] = ordered-max
__global__ void init_ws(u32* ws) {
  int i = threadIdx.x;
  if (i < 4) ws[i] = 0xFFFFFFFFu;      // min identity
  else if (i < 8) ws[i] = 0u;          // max identity
}

__global__ __launch_bounds__(THREADS)
void reduce_minmax(const float* __restrict__ coords, int N, u32* ws) {
  __shared__ u32 smin[4], smax[4];
  if (threadIdx.x < 4) { smin[threadIdx.x] = 0xFFFFFFFFu; smax[threadIdx.x] = 0u; }
  __syncthreads();

  const float4* c4 = (const float4*)coords;
  float lmin[4] = { INFINITY,  INFINITY,  INFINITY,  INFINITY};
  float lmax[4] = {-INFINITY, -INFINITY, -INFINITY, -INFINITY};
  int stride = gridDim.x * blockDim.x;
  for (int p = blockIdx.x * blockDim.x + threadIdx.x; p < N; p += stride) {
    float4 c = c4[p];                               // global_load_b128
    float cx, cy, cz;
    ecef(c.x, c.y, c.z, cx, cy, cz);
    float v[4] = {cx, cy, cz, c.w};
#pragma unroll
    for (int i = 0; i < 4; ++i) {
      lmin[i] = fminf(lmin[i], v[i]);
      lmax[i] = fmaxf(lmax[i], v[i]);
    }
  }
#pragma unroll
  for (int i = 0; i < 4; ++i) {
    atomicMin(&smin[i], f2ord(lmin[i]));
    atomicMax(&smax[i], f2ord(lmax[i]));
  }
  __syncthreads();
  if (threadIdx.x < 4) {
    atomicMin(&ws[threadIdx.x], smin[threadIdx.x]);
    atomicMax(&ws[threadIdx.x + 4], smax[threadIdx.x]);
  }
}

__global__ __launch_bounds__(THREADS)
void encode_kernel(const float* __restrict__ coords,
                   const float* __restrict__ t_xyz,
                   const float* __restrict__ t_xyt,
                   const float* __restrict__ t_yzt,
                   const float* __restrict__ t_xzt,
                   const u32* __restrict__ ws,
                   float* __restrict__ out, int N, ResConst rc) {
  __shared__ __align__(16) float sCoords[PTS_PER_BLK * 4]; // 1 KB, via TDM
  __shared__ __align__(16) float sOut[64 * SROW];  // 32.5 KB staging

  int blockBase = blockIdx.x * PTS_PER_BLK;

  // --- TDM: async-load this block's coords tile (zero-fills OOB tail) ------
  if (threadIdx.x < 32) {  // wave 0 only (TDM ignores EXEC; wave-level issue)
    u32 ldsAddr = (u32)(uintptr_t)(&sCoords[0]);   // low 32b of flat = LDS off
    unsigned long long ga =
        (unsigned long long)(uintptr_t)(coords + (size_t)blockBase * 4);
    u32 dim0 = (u32)((long long)N * 4 - (long long)blockBase * 4);

    u32x4 g0;
    g0.x = 1u;                                      // count=1, user descriptor
    g0.y = ldsAddr;                                 // lds_addr (bytes)
    g0.z = (u32)(ga & 0xFFFFFFFFull);               // global_addr[31:0]
    g0.w = (u32)((ga >> 32) & 0x1FFFFFFull) | (2u << 30); // [56:32] | type=2

    i32x8 g1 = {0, 0, 0, 0, 0, 0, 0, 0};
    g1[0] = (int)(2u << 16);                        // data_size = 4B
    g1[1] = (int)((dim0 & 0xFFFFu) << 16);          // tensor_dim0[15:0]
    g1[2] = (int)(((dim0 >> 16) & 0xFFFFu) | (1u << 16)); // dim0 hi | dim1=1
    g1[3] = (int)((u32)(PTS_PER_BLK * 4) << 16);    // tile_dim0 = 256 elems
    g1[4] = 1;                                      // tile_dim1 = 1
    g1[5] = (int)dim0;                              // dim0_stride (unused, 1D)
    i32x4 gz = {0, 0, 0, 0};
#if defined(__clang_major__) && (__clang_major__ >= 23)
    i32x8 gz8 = {0, 0, 0, 0, 0, 0, 0, 0};
    __builtin_amdgcn_tensor_load_to_lds(g0, g1, gz, gz, gz8, 0);
#else
    __builtin_amdgcn_tensor_load_to_lds(g0, g1, gz, gz, 0);
#endif
    __builtin_amdgcn_s_wait_tensorcnt(0);
  }
  __syncthreads();

  // --- wave-uniform mapping: enc = wave&3 (scalar table base / resolutions),
  //     point = (wave>>2)*32 + lane -----------------------------------------
  int wave = threadIdx.x >> 5;
  int lane = threadIdx.x & 31;
  int enc  = wave & 3;                              // uniform within wave
  int lp   = (wave >> 2) * 32 + lane;               // local point 0..63

  float4 c = ((const float4*)sCoords)[lp];          // ds_load_b128
  float cx, cy, cz;
  ecef(c.x, c.y, c.z, cx, cy, cz);

  float mn0 = ord2f(ws[0]), mn1 = ord2f(ws[1]), mn2 = ord2f(ws[2]), mnt = ord2f(ws[3]);
  float mx0 = ord2f(ws[4]), mx1 = ord2f(ws[5]), mx2 = ord2f(ws[6]), mxt = ord2f(ws[7]);

  float xn = (cx - mn0) / (mx0 - mn0);
  float yn = (cy - mn1) / (mx1 - mn1);
  float zn = (cz - mn2) / (mx2 - mn2);
  float tn = (c.w - mnt) / (mxt - mnt);
  float ts = (tn * 2.0f - 1.0f) * 0.9f;

  float vx, vy, vz;
  const float* tab;
  switch (enc) {                                    // uniform scalar branch
    case 0:  vx = xn; vy = yn; vz = zn; tab = t_xyz; break;
    case 1:  vx = xn; vy = yn; vz = ts; tab = t_xyt; break;
    case 2:  vx = yn; vy = zn; vz = ts; tab = t_yzt; break;
    default: vx = xn; vy = zn; vz = ts; tab = t_xzt; break;
  }
  float xs = fminf(fmaxf((vx + 1.0f) * 0.5f, 0.0f), 1.0f);
  float ys = fminf(fmaxf((vy + 1.0f) * 0.5f, 0.0f), 1.0f);
  float zs = fminf(fmaxf((vz + 1.0f) * 0.5f, 0.0f), 1.0f);

#pragma unroll
  for (int l = 0; l < LVLS; ++l) {
    int rx, ry, rz;
    if (enc == 0) { rx = ry = rz = rc.sres[l]; }    // uniform scalar select
    else          { rx = ry = rc.tres01[l]; rz = rc.tres2[l]; }

    float px = xs * ((float)rx - 1.0f);
    float py = ys * ((float)ry - 1.0f);
    float pz = zs * ((float)rz - 1.0f);
    int p0x = (int)floorf(px), p0y = (int)floorf(py), p0z = (int)floorf(pz);
    float wx1 = px - (float)p0x, wy1 = py - (float)p0y, wz1 = pz - (float)p0z;
    float wx0 = 1.0f - wx1, wy0 = 1.0f - wy1, wz0 = 1.0f - wz1;

    // Factored hash bases: p0 <= r-1 always, only +1 corners need the clamp.
    u32 hx[2], hyz[4];
    hx[0] = (u32)p0x;
    hx[1] = (u32)min(p0x + 1, rx - 1);
    u32 hy0 = (u32)p0y * 2654435761u;
    u32 hy1 = (u32)min(p0y + 1, ry - 1) * 2654435761u;
    u32 hz0 = (u32)p0z * 805459861u;
    u32 hz1 = (u32)min(p0z + 1, rz - 1) * 805459861u;
    hyz[0] = hy0 ^ hz0; hyz[1] = hy1 ^ hz0;
    hyz[2] = hy0 ^ hz1; hyz[3] = hy1 ^ hz1;

    float wxv[2] = {wx0, wx1};
    float wyz[4] = {wy0 * wz0, wy1 * wz0, wy0 * wz1, wy1 * wz1};

    const float* ltab = tab + (size_t)l * (TBL * 2); // scalar per-level base
    float a0 = 0.0f, a1 = 0.0f;
#pragma unroll
    for (int cr = 0; cr < 8; ++cr) {
      u32 idx = (hx[cr & 1] ^ hyz[cr >> 1]) & TMASK;
      const float2 f = *(const float2*)(ltab + (size_t)idx * 2); // b64 gather
      float wc = wxv[cr & 1] * wyz[cr >> 1];
      a0 += f.x * wc;
      a1 += f.y * wc;
    }
    // One b64 store per level; row = enc*16+l, lanes cover all 64 banks once.
    *(float2*)&sOut[(enc * 16 + l) * SROW + lp * 2] = make_float2(a0, a1);
  }
  __syncthreads();

  // --- coalesced b128 writeback of valid rows (128 floats/point) -----------
  int valid = N - blockBase;
  if (valid > PTS_PER_BLK) valid = PTS_PER_BLK;
  int nvec = valid * 32;                            // float4 count
  float4* out4 = (float4*)(out + (size_t)blockBase * 128);
  for (int v = threadIdx.x; v < nvec; v += THREADS) {
    int i = v * 4;
    int p = i >> 7;                                 // point row
    int r = (i & 127) >> 1;                         // sOut row (even)
    float2 lo = *(const float2*)&sOut[(r + 0) * SROW + p * 2];
    float2 hi = *(const float2*)&sOut[(r + 1) * SROW + p * 2];
    float4 val = make_float4(lo.x, lo.y, hi.x, hi.y);
    out4[v] = val;                                  // global_store_b128
  }
}

// ---------------------------------------------------------------------------
extern "C" void kernel_launch(void* const* d_in, const int* in_sizes, int n_in,
                              void* d_out, int out_size, void* d_ws, size_t ws_size,
                              hipStream_t stream) {
  const float* coords = (const float*)d_in[0];
  const float* t_xyz  = (const float*)d_in[1];
  const float* t_xyt  = (const float*)d_in[2];
  const float* t_yzt  = (const float*)d_in[3];
  const float* t_xzt  = (const float*)d_in[4];
  float* out = (float*)d_out;
  u32* ws = (u32*)d_ws;
  int N = in_sizes[0] / 4;

  // Replicate numpy _level_resolutions exactly in double precision.
  ResConst rc;
  {
    double sS  = std::exp((std::log(512.0) - std::log(16.0)) / 15.0);
    double sT0 = std::exp((std::log(32.0)  - std::log(8.0))  / 15.0);
    double sT2 = std::exp((std::log(16.0)  - std::log(8.0))  / 15.0);
    for (int l = 0; l < LVLS; ++l) {
      rc.sres[l]   = (int)std::floor(16.0 * std::pow(sS,  (double)l) + 0.5);
      rc.tres01[l] = (int)std::floor( 8.0 * std::pow(sT0, (double)l) + 0.5);
      rc.tres2[l]  = (int)std::floor( 8.0 * std::pow(sT2, (double)l) + 0.5);
    }
  }

  init_ws<<<1, 32, 0, stream>>>(ws);

  int rblocks = (N + THREADS - 1) / THREADS;
  if (rblocks > 512) rblocks = 512;
  reduce_minmax<<<rblocks, THREADS, 0, stream>>>(coords, N, ws);

  int eblocks = (N + PTS_PER_BLK - 1) / PTS_PER_BLK;
  encode_kernel<<<eblocks, THREADS, 0, stream>>>(coords, t_xyz, t_xyt, t_yzt,
                                                 t_xzt, ws, out, N, rc);
}